// GraphSNN_84799834293182
// MI455X (gfx1250) — compile-verified
//
#include <hip/hip_runtime.h>

// fp32 WMMA fragments for V_WMMA_F32_16X16X4_F32 (wave32):
//   A: 16x4 f32 -> 2 VGPRs, B: 4x16 f32 -> 2 VGPRs, C/D: 16x16 f32 -> 8 VGPRs
typedef __attribute__((ext_vector_type(2))) float v2f;
typedef __attribute__((ext_vector_type(8))) float v8f;

#define TILE_LD 17  // padded LDS row stride (17 coprime with 64 banks)

__device__ __forceinline__ v8f wmma4(v2f a, v2f b, v8f c) {
  return __builtin_amdgcn_wmma_f32_16x16x4_f32(false, a, false, b, (short)0, c,
                                               false, false);
}

// One dense layer on a 16-row activation tile in this wave's LDS tile.
// tile[m*TILE_LD + k] = activation(row m, feature k). W is Kdim x Ndim
// row-major. Kdim/Ndim compile-time so zero-slices are dropped entirely.
template <int Kdim, int Ndim>
__device__ __forceinline__ v8f wmma_layer_lds(const float* tile,
                                              const float* __restrict__ W,
                                              const float* __restrict__ bias,
                                              int lane) {
  static_assert(Kdim % 4 == 0 && Kdim <= 16 && (Ndim & (Ndim - 1)) == 0, "");
  constexpr int KK = Kdim / 4;  // #WMMA slices that are not all-zero padding
  const int n = lane & 15, hi = lane >> 4, m = lane & 15;
  v2f a[KK], b[KK];
#pragma unroll
  for (int kk = 0; kk < KK; ++kk) {
    const int k0 = kk * 4 + hi * 2;  // ISA A/B layout: vgpr0=K{k0}, vgpr1=K{k0+1}
    a[kk].x = tile[m * TILE_LD + k0];
    a[kk].y = tile[m * TILE_LD + k0 + 1];
    // branch-free N padding: clamped load + select (no exec manipulation)
    const float w0v = W[(k0    ) * Ndim + (n & (Ndim - 1))];
    const float w1v = W[(k0 + 1) * Ndim + (n & (Ndim - 1))];
    b[kk].x = (n < Ndim) ? w0v : 0.0f;
    b[kk].y = (n < Ndim) ? w1v : 0.0f;
  }
  v8f c = {};
#pragma unroll
  for (int kk = 0; kk < KK; ++kk) c = wmma4(a[kk], b[kk], c);
  const float bn = (n < Ndim) ? bias[n & (Ndim - 1)] : 0.0f;
#pragma unroll
  for (int i = 0; i < 8; ++i) c[i] = fmaxf(c[i] + bn, 0.0f);
  return c;
}

// Spill a C/D-layout accumulator into the wave's LDS tile (row-major, padded).
__device__ __forceinline__ void store_tile(float* tile, v8f c, int lane) {
  const int n     = lane & 15;
  const int mbase = (lane >> 4) * 8;  // C/D layout: lane half selects M+8
#pragma unroll
  for (int i = 0; i < 8; ++i) tile[(mbase + i) * TILE_LD + n] = c[i];
}

// ---------------- per-node MLP + segment-sum into dag_summ ----------------
__global__ void __launch_bounds__(256)
node_kernel(const float* __restrict__ inputs,        // (N, 13)
            const int* __restrict__ node_to_dag,     // (N,)
            const float* __restrict__ w0, const float* __restrict__ b0,  // 13x16
            const float* __restrict__ w1, const float* __restrict__ b1,  // 16x8
            const float* __restrict__ w2, const float* __restrict__ b2,  // 8x8
            float* __restrict__ dag_summ,            // (D, 8), pre-zeroed
            int n_nodes) {
  __shared__ float tiles[8][16 * TILE_LD];
  const int wave = __builtin_amdgcn_readfirstlane(threadIdx.x >> 5);  // scalar
  const int lane = threadIdx.x & 31;
  float* tile = tiles[wave];

  const int node0 = (blockIdx.x * 8 + wave) * 16;  // uniform per wave
  const int n  = lane & 15;
  const int hi = lane >> 4;
  const int m  = lane & 15;

  // ---- layer 0 A: 16 nodes x 13 features, K padded to 16 ----
  v2f a[4];
  if (node0 + 16 <= n_nodes) {            // scalar branch: hot path unguarded
    const int nd13 = (node0 + m) * 13;
#pragma unroll
    for (int kk = 0; kk < 3; ++kk) {
      const int k0 = kk * 4 + hi * 2;     // k0 <= 10, k0+1 <= 11: in-bounds
      a[kk].x = inputs[nd13 + k0];
      a[kk].y = inputs[nd13 + k0 + 1];
    }
    const float t = inputs[nd13 + 12];    // only K=12 survives in last slice
    a[3].x = hi ? 0.0f : t;
    a[3].y = 0.0f;
  } else {                                // cold tail: masked per element
    const int nd = node0 + m;
#pragma unroll
    for (int kk = 0; kk < 4; ++kk) {
      const int k0 = kk * 4 + hi * 2;
      a[kk].x = (nd < n_nodes && k0     < 13) ? inputs[nd * 13 + k0    ] : 0.0f;
      a[kk].y = (nd < n_nodes && k0 + 1 < 13) ? inputs[nd * 13 + k0 + 1] : 0.0f;
    }
  }
  // B: w0 is 13x16 (full N) — same K=12 tail handling, branch-free
  v2f b[4];
#pragma unroll
  for (int kk = 0; kk < 3; ++kk) {
    const int k0 = kk * 4 + hi * 2;
    b[kk].x = w0[(k0    ) * 16 + n];
    b[kk].y = w0[(k0 + 1) * 16 + n];
  }
  {
    const float wt = w0[12 * 16 + n];
    b[3].x = hi ? 0.0f : wt;
    b[3].y = 0.0f;
  }
  v8f c = {};
#pragma unroll
  for (int kk = 0; kk < 4; ++kk) c = wmma4(a[kk], b[kk], c);
#pragma unroll
  for (int i = 0; i < 8; ++i) c[i] = fmaxf(c[i] + b0[n], 0.0f);

  store_tile(tile, c, lane);
  __syncthreads();
  v8f c1 = wmma_layer_lds<16, 8>(tile, w1, b1, lane);   // 16 -> 8, 4 WMMAs
  __syncthreads();
  store_tile(tile, c1, lane);                           // cols 8..15 are 0
  __syncthreads();
  v8f c2 = wmma_layer_lds<8, 8>(tile, w2, b2, lane);    // 8 -> 8, 2 WMMAs

  // ---- segment sum: scatter 16x8 result with f32 L2 atomics ----
  if (n < 8) {
    const int mbase = hi * 8;
#pragma unroll
    for (int i = 0; i < 8; ++i) {
      const int nd = node0 + mbase + i;
      if (nd < n_nodes)
        atomicAdd(&dag_summ[node_to_dag[nd] * 8 + n], c2[i]);
    }
  }
}

// ---------------- per-dag MLP + global reduction ----------------
__global__ void __launch_bounds__(256)
dag_kernel(const float* __restrict__ dag_summ,       // (D, 8)
           const float* __restrict__ w0, const float* __restrict__ b0,  // 8x16
           const float* __restrict__ w1, const float* __restrict__ b1,  // 16x8
           const float* __restrict__ w2, const float* __restrict__ b2,  // 8x8
           float* __restrict__ global_out,           // (8,), pre-zeroed
           int num_dags) {
  __shared__ float tiles[8][16 * TILE_LD];
  __shared__ float acc[8];
  const int wave = __builtin_amdgcn_readfirstlane(threadIdx.x >> 5);
  const int lane = threadIdx.x & 31;
  float* tile = tiles[wave];

  if (threadIdx.x < 8) acc[threadIdx.x] = 0.0f;
  __syncthreads();

  const int dag0 = (blockIdx.x * 8 + wave) * 16;
  const int n  = lane & 15;
  const int hi = lane >> 4;
  const int m  = lane & 15;

  // ---- layer 0 A: 16 dags x 8, K padded to 16 -> only 2 live WMMA slices ----
  v2f a0, a1;
  if (dag0 + 16 <= num_dags) {            // hot path: 8B-aligned v2f loads
    const float* row = dag_summ + (dag0 + m) * 8;
    a0 = *(const v2f*)(row + 2 * hi);         // K = {2hi, 2hi+1}
    a1 = *(const v2f*)(row + 4 + 2 * hi);     // K = {4+2hi, 5+2hi}
  } else {
    const int dg = dag0 + m;
    a0.x = (dg < num_dags) ? dag_summ[dg * 8 + 2 * hi    ] : 0.0f;
    a0.y = (dg < num_dags) ? dag_summ[dg * 8 + 2 * hi + 1] : 0.0f;
    a1.x = (dg < num_dags) ? dag_summ[dg * 8 + 2 * hi + 4] : 0.0f;
    a1.y = (dg < num_dags) ? dag_summ[dg * 8 + 2 * hi + 5] : 0.0f;
  }
  v2f b0f, b1f;  // glob_w0 is 8x16, all K in-bounds for the 2 live slices
  b0f.x = w0[(2 * hi    ) * 16 + n];
  b0f.y = w0[(2 * hi + 1) * 16 + n];
  b1f.x = w0[(2 * hi + 4) * 16 + n];
  b1f.y = w0[(2 * hi + 5) * 16 + n];

  v8f c = {};
  c = wmma4(a0, b0f, c);
  c = wmma4(a1, b1f, c);
#pragma unroll
  for (int i = 0; i < 8; ++i) c[i] = fmaxf(c[i] + b0[n], 0.0f);

  store_tile(tile, c, lane);
  __syncthreads();
  v8f c1 = wmma_layer_lds<16, 8>(tile, w1, b1, lane);   // 4 WMMAs
  __syncthreads();
  store_tile(tile, c1, lane);
  __syncthreads();
  v8f c2 = wmma_layer_lds<8, 8>(tile, w2, b2, lane);    // 2 WMMAs

  // ---- reduce over dags: LDS accumulate, then 8 global atomics per block ----
  if (n < 8) {
    const int mbase = hi * 8;
    float partial = 0.0f;
#pragma unroll
    for (int i = 0; i < 8; ++i) {
      const int dg = dag0 + mbase + i;
      if (dg < num_dags) partial += c2[i];
    }
    atomicAdd(&acc[n], partial);
  }
  __syncthreads();
  if (threadIdx.x < 8) atomicAdd(&global_out[threadIdx.x], acc[threadIdx.x]);
}

extern "C" void kernel_launch(void* const* d_in, const int* in_sizes, int n_in,
                              void* d_out, int out_size, void* d_ws, size_t ws_size,
                              hipStream_t stream) {
  // setup_inputs() order:
  // 0 inputs, 1 node_to_dag, 2 num_dags, 3..8 dag_{w0,b0,w1,b1,w2,b2},
  // 9..14 glob_{w0,b0,w1,b1,w2,b2}
  const float* inputs      = (const float*)d_in[0];
  const int*   node_to_dag = (const int*)d_in[1];
  const float* dag_w0 = (const float*)d_in[3];
  const float* dag_b0 = (const float*)d_in[4];
  const float* dag_w1 = (const float*)d_in[5];
  const float* dag_b1 = (const float*)d_in[6];
  const float* dag_w2 = (const float*)d_in[7];
  const float* dag_b2 = (const float*)d_in[8];
  const float* glob_w0 = (const float*)d_in[9];
  const float* glob_b0 = (const float*)d_in[10];
  const float* glob_w1 = (const float*)d_in[11];
  const float* glob_b1 = (const float*)d_in[12];
  const float* glob_w2 = (const float*)d_in[13];
  const float* glob_b2 = (const float*)d_in[14];

  const int n_nodes  = in_sizes[0] / 13;
  const int num_dags = (out_size - 8) / 8;   // d_out = dag_summ (D*8) ++ global (8)

  float* dag_summ   = (float*)d_out;
  float* global_out = dag_summ + (size_t)num_dags * 8;

  // Both kernels accumulate atomically -> zero the output first (capturable).
  hipMemsetAsync(d_out, 0, (size_t)out_size * sizeof(float), stream);

  const int node_blocks = (n_nodes + 127) / 128;   // 8 waves x 16 nodes / block
  node_kernel<<<node_blocks, 256, 0, stream>>>(inputs, node_to_dag,
                                               dag_w0, dag_b0, dag_w1, dag_b1,
                                               dag_w2, dag_b2,
                                               dag_summ, n_nodes);

  const int dag_blocks = (num_dags + 127) / 128;   // 8 waves x 16 dags / block
  dag_kernel<<<dag_blocks, 256, 0, stream>>>(dag_summ,
                                             glob_w0, glob_b0, glob_w1, glob_b1,
                                             glob_w2, glob_b2,
                                             global_out, num_dags);
}